// RobustTripletLoss_final_5368709120079
// MI455X (gfx1250) — compile-verified
//
#include <hip/hip_runtime.h>
#include <hip/hip_bf16.h>
#include <stdint.h>
#include <math.h>

#define NROWS 4096
#define NDIM  1024
#define NCLS  64
#define KSEL  20           // top-K for mean_topk
#define KNEG  12           // hn + 11 candidates
#define MARGIN 0.3f
#define MT    128          // 32x32 macro-tiles per dimension

typedef __attribute__((ext_vector_type(2)))  float    v2f;
typedef __attribute__((ext_vector_type(8)))  float    v8f;

// ---------------- helpers ----------------

__device__ __forceinline__ float reluf(float x) { return fmaxf(x, 0.0f); }

// deterministic gumbel noise: splitmix64 hash -> (0,1) -> -log(-log(u))
__device__ __forceinline__ float gumbel_rand(uint32_t stream, uint32_t i, uint32_t j) {
  uint64_t z = ((uint64_t)i << 32) | (uint64_t)j;
  z ^= (uint64_t)(stream + 1u) * 0xD1B54A32D192ED03ULL;
  z += 0x9E3779B97F4A7C15ULL;
  z = (z ^ (z >> 30)) * 0xBF58476D1CE4E5B9ULL;
  z = (z ^ (z >> 27)) * 0x94D049BB133111EBULL;
  z ^= (z >> 31);
  uint32_t m = (uint32_t)(z >> 40);                   // 24 random bits
  float u = ((float)m + 0.5f) * (1.0f / 16777216.0f); // in (0,1)
  return -__logf(-__logf(u));
}

// keep K smallest, sorted by (value asc, index asc) -- matches stable argsort
template <int K>
__device__ __forceinline__ void ins_asc(float* v, int* id, float nv, int nj) {
  float lv = v[K - 1]; int li = id[K - 1];
  if (nv > lv || (nv == lv && nj > li)) return;
  int p = K - 1;
  while (p > 0 && (nv < v[p - 1] || (nv == v[p - 1] && nj < id[p - 1]))) {
    v[p] = v[p - 1]; id[p] = id[p - 1]; --p;
  }
  v[p] = nv; id[p] = nj;
}

// keep K largest, sorted by (value desc, index asc) -- matches argsort(-x)
template <int K>
__device__ __forceinline__ void ins_desc(float* v, int* id, float nv, int nj) {
  float lv = v[K - 1]; int li = id[K - 1];
  if (nv < lv || (nv == lv && nj > li)) return;
  int p = K - 1;
  while (p > 0 && (nv > v[p - 1] || (nv == v[p - 1] && nj < id[p - 1]))) {
    v[p] = v[p - 1]; id[p] = id[p - 1]; --p;
  }
  v[p] = nv; id[p] = nj;
}

// ---------------- kernel 1: row squared norms ----------------

__global__ void __launch_bounds__(256) k_rowsq(const float* __restrict__ X,
                                               float* __restrict__ sq) {
  int wave = threadIdx.x >> 5, lane = threadIdx.x & 31;
  int row  = blockIdx.x * 8 + wave;
  const float* xr = X + (size_t)row * NDIM;
  float s = 0.f;
  for (int k = lane; k < NDIM; k += 32) { float v = xr[k]; s = fmaf(v, v, s); }
  #pragma unroll
  for (int m = 16; m >= 1; m >>= 1) s += __shfl_xor(s, m, 32);
  if (lane == 0) sq[row] = s;
}

// ---------------- kernel 2: prediction argmax (first max wins) ----------------

__global__ void __launch_bounds__(256) k_predcls(const float* __restrict__ P,
                                                 int* __restrict__ cls) {
  int r = blockIdx.x * blockDim.x + threadIdx.x;
  if (r >= NROWS) return;
  const float* p = P + (size_t)r * NCLS;
  float best = p[0]; int bi = 0;
  for (int c = 1; c < NCLS; ++c) { float v = p[c]; if (v > best) { best = v; bi = c; } }
  cls[r] = bi;
}

// ---------------- kernel 3: pairwise distance via f32 WMMA ----------------
// One wave per 32x32 macro-tile (2x2 grid of 16x16 WMMA tiles): 1 b64 load
// per v_wmma instead of 2, halving L2 traffic per FLOP. Triangular macro-tile
// grid (tj >= ti) with mirrored stores halves the GEMM via symmetry.

__global__ void __launch_bounds__(32) k_dist(const float* __restrict__ X,
                                             const float* __restrict__ sq,
                                             float* __restrict__ dist) {
  int L = (int)blockIdx.x;
  // invert cumulative base(t) = MT*t - t*(t-1)/2
  float f = (MT + 0.5f) - sqrtf((MT + 0.5f) * (MT + 0.5f) - 2.0f * (float)L);
  int ti = (int)f; if (ti < 0) ti = 0; if (ti > MT - 1) ti = MT - 1;
  while (ti < MT - 1 && (MT * (ti + 1) - ((ti + 1) * ti) / 2) <= L) ++ti;
  while (ti > 0      && (MT * ti - (ti * (ti - 1)) / 2) > L) --ti;
  int tj = ti + (L - (MT * ti - (ti * (ti - 1)) / 2));
  int i0 = ti * 32, j0 = tj * 32;

  int lane = (int)threadIdx.x;
  int half = lane >> 4;     // 0: K 0,1   1: K 2,3  (f32 A/B layout)
  int l    = lane & 15;

  const float* pa0 = X + (size_t)(i0 + l)      * NDIM + 2 * half;
  const float* pa1 = X + (size_t)(i0 + 16 + l) * NDIM + 2 * half;
  const float* pb0 = X + (size_t)(j0 + l)      * NDIM + 2 * half;
  const float* pb1 = X + (size_t)(j0 + 16 + l) * NDIM + 2 * half;

  v8f acc00 = {}, acc01 = {}, acc10 = {}, acc11 = {};
  #pragma unroll 2
  for (int kk = 0; kk < NDIM; kk += 4) {
    float2 a0f = *(const float2*)(pa0 + kk);
    float2 a1f = *(const float2*)(pa1 + kk);
    float2 b0f = *(const float2*)(pb0 + kk);
    float2 b1f = *(const float2*)(pb1 + kk);
    v2f a0; a0[0] = a0f.x; a0[1] = a0f.y;
    v2f a1; a1[0] = a1f.x; a1[1] = a1f.y;
    v2f b0; b0[0] = b0f.x; b0[1] = b0f.y;
    v2f b1; b1[0] = b1f.x; b1[1] = b1f.y;
    // 4 independent accumulators: no D->A/B RAW between consecutive WMMAs
    acc00 = __builtin_amdgcn_wmma_f32_16x16x4_f32(false, a0, false, b0, (short)0, acc00, false, false);
    acc01 = __builtin_amdgcn_wmma_f32_16x16x4_f32(false, a0, false, b1, (short)0, acc01, false, false);
    acc10 = __builtin_amdgcn_wmma_f32_16x16x4_f32(false, a1, false, b0, (short)0, acc10, false, false);
    acc11 = __builtin_amdgcn_wmma_f32_16x16x4_f32(false, a1, false, b1, (short)0, acc11, false, false);
  }

  // C/D layout: VGPR v, lanes 0-15 -> M=v, lanes 16-31 -> M=v+8; N = lane&15
  bool mirror = (ti != tj);
  #pragma unroll
  for (int st = 0; st < 4; ++st) {
    const v8f& acc = (st == 0) ? acc00 : (st == 1) ? acc01 : (st == 2) ? acc10 : acc11;
    int rbase = i0 + ((st >> 1) ? 16 : 0) + 8 * half;
    int col   = j0 + ((st & 1) ? 16 : 0) + l;
    float sqc = sq[col];
    #pragma unroll
    for (int v = 0; v < 8; ++v) {
      int row  = rbase + v;
      float d2 = sq[row] + sqc - 2.0f * acc[v];
      float d  = sqrtf(fmaxf(d2, 1e-12f));
      dist[(size_t)row * NROWS + col] = d;
      if (mirror) dist[(size_t)col * NROWS + row] = d;
    }
  }
}

// ---------------- kernel 4: per-row mining + loss term ----------------
// One wave per row; 8 waves per block. LDS used for top-k merge.

__global__ void __launch_bounds__(256) k_mine(const float* __restrict__ dist,
                                              const int* __restrict__ targets,
                                              const float* __restrict__ prob,
                                              const int* __restrict__ predcls,
                                              const float* __restrict__ thrp,
                                              float* __restrict__ cArr,
                                              int* __restrict__ confArr,
                                              int* __restrict__ corrArr) {
  __shared__ float ssv[8][32 * KSEL];
  __shared__ int   ssi[8][32 * KSEL];

  int w = threadIdx.x >> 5, lane = threadIdx.x & 31;
  int i = blockIdx.x * 8 + w;
  const float* drow = dist + (size_t)i * NROWS;
  float thr  = *thrp;
  int   clsi = targets[i];

  // ---- phase 1: top-12 smallest negatives ----
  float nv[KNEG]; int ni[KNEG];
  #pragma unroll
  for (int t = 0; t < KNEG; ++t) { nv[t] = INFINITY; ni[t] = 0x7FFFFFFF; }
  for (int j = lane; j < NROWS; j += 32)
    if (targets[j] != clsi) ins_asc<KNEG>(nv, ni, drow[j], j);
  #pragma unroll
  for (int t = 0; t < KNEG; ++t) { ssv[w][lane * KNEG + t] = nv[t]; ssi[w][lane * KNEG + t] = ni[t]; }
  __syncthreads();

  int hn = 0, hn_new = 0;
  if (lane == 0) {
    float mv[KNEG]; int mi[KNEG];
    #pragma unroll
    for (int t = 0; t < KNEG; ++t) { mv[t] = INFINITY; mi[t] = 0x7FFFFFFF; }
    for (int e = 0; e < 32 * KNEG; ++e) ins_asc<KNEG>(mv, mi, ssv[w][e], ssi[w][e]);
    hn = mi[0];
    hn_new = mi[KNEG - 1];                 // last candidate forced "confident"
    for (int k = 1; k < KNEG; ++k) {
      int cj = mi[k];
      if (k == KNEG - 1 || prob[cj] >= thr) { hn_new = cj; break; }
    }
  }
  __syncthreads();

  // ---- phase 2: top-20 largest of cols 64..4095 ("bugged" gather at idx-64) ----
  float tv[KSEL]; int tix[KSEL];
  #pragma unroll
  for (int t = 0; t < KSEL; ++t) { tv[t] = -INFINITY; tix[t] = 0x7FFFFFFF; }
  for (int j = 64 + lane; j < NROWS; j += 32) ins_desc<KSEL>(tv, tix, drow[j], j);
  #pragma unroll
  for (int t = 0; t < KSEL; ++t) { ssv[w][lane * KSEL + t] = tv[t]; ssi[w][lane * KSEL + t] = tix[t]; }
  __syncthreads();

  float IC = 0.f, VA = 0.f, VB = 0.f;
  if (lane == 0) {
    float mv[KSEL]; int mi[KSEL];
    #pragma unroll
    for (int t = 0; t < KSEL; ++t) { mv[t] = -INFINITY; mi[t] = 0x7FFFFFFF; }
    for (int e = 0; e < 32 * KSEL; ++e) ins_desc<KSEL>(mv, mi, ssv[w][e], ssi[w][e]);
    float s = 0.f;
    for (int t = 0; t < KSEL; ++t) s += drow[mi[t] - 64];   // bug: slice idx into full cols
    IC = s / (float)KSEL;

    // VA: top-20 of cols 0..31 (gather is identity there)
    float av[KSEL]; int ai[KSEL];
    #pragma unroll
    for (int t = 0; t < KSEL; ++t) { av[t] = -INFINITY; ai[t] = 0x7FFFFFFF; }
    for (int j = 0; j < 32; ++j) ins_desc<KSEL>(av, ai, drow[j], j);
    s = 0.f; for (int t = 0; t < KSEL; ++t) s += av[t];
    VA = s / (float)KSEL;

    // VB: top-20 of cols 32..63, gather at idx-32
    float bv[KSEL]; int bi2[KSEL];
    #pragma unroll
    for (int t = 0; t < KSEL; ++t) { bv[t] = -INFINITY; bi2[t] = 0x7FFFFFFF; }
    for (int j = 32; j < 64; ++j) ins_desc<KSEL>(bv, bi2, drow[j], j);
    s = 0.f; for (int t = 0; t < KSEL; ++t) s += drow[bi2[t] - 32];
    VB = s / (float)KSEL;
  }

  // ---- phase 3: mask fallbacks ----
  bool a1l = false, a2l = false;
  for (int j = lane; j < NROWS; j += 32) {
    bool pos = (targets[j] == clsi);
    bool m1  = pos && (j != i);
    a1l |= m1;
    a2l |= (m1 && (prob[j] >= thr));
  }
  bool any1 = (__ballot(a1l) != 0ull);
  bool any2 = (__ballot(a2l) != 0ull);

  // ---- phase 4: gumbel argmax for p_idx (g1/maskP) and p_new (g2/maskP2) ----
  float g1b = -INFINITY; int p1 = 0;
  float g2b = -INFINITY; int p2 = 0;
  for (int j = lane; j < NROWS; j += 32) {
    bool pos    = (targets[j] == clsi);
    bool m1     = pos && (j != i);
    bool maskP  = any1 ? m1 : pos;
    bool m2     = m1 && (prob[j] >= thr);
    bool maskP2 = any2 ? m2 : maskP;
    if (maskP)  { float g = gumbel_rand(0u, (uint32_t)i, (uint32_t)j);
                  if (g > g1b) { g1b = g; p1 = j; } }
    if (maskP2) { float g = gumbel_rand(1u, (uint32_t)i, (uint32_t)j);
                  if (g > g2b) { g2b = g; p2 = j; } }
  }
  #pragma unroll
  for (int m = 16; m >= 1; m >>= 1) {
    float ov = __shfl_xor(g1b, m, 32); int oi = __shfl_xor(p1, m, 32);
    if (ov > g1b || (ov == g1b && oi < p1)) { g1b = ov; p1 = oi; }
    float ov2 = __shfl_xor(g2b, m, 32); int oi2 = __shfl_xor(p2, m, 32);
    if (ov2 > g2b || (ov2 == g2b && oi2 < p2)) { g2b = ov2; p2 = oi2; }
  }

  // ---- phase 5: assemble per-row loss term ----
  if (lane == 0) {
    float d_ip      = drow[p1];
    float d_ihn     = drow[hn];
    float d_ihn_new = drow[hn_new];
    float d_ipnew   = drow[p2];
    bool conf_p  = prob[p1] >= thr;
    bool conf_hn = prob[hn] >= thr;
    bool FN      = (predcls[hn] == clsi);

    float ap = d_ip, an = d_ihn;
    bool bFN = conf_p && !conf_hn && FN;
    if (bFN)  { ap = (d_ip + d_ihn) * 0.5f; an = d_ihn_new; }
    bool cRes = (!conf_p && conf_hn) || (!conf_p && !conf_hn && !FN);
    if (cRes) { an = (d_ip + d_ihn) * 0.5f; ap = d_ipnew; }
    bool inv  = !conf_p && !conf_hn && FN;
    float c_conf = inv ? reluf(an - ap + MARGIN) : reluf(ap - an + MARGIN);

    float apA = drow[(i + 32 < NROWS) ? (i + 32) : (NROWS - 1)];
    float apB = drow[(i - 32 > 0) ? (i - 32) : 0];
    float apC = drow[i];
    float cA = (reluf(apA - IC + MARGIN) + reluf(apA - VA + MARGIN)) * 0.5f;
    float cB = (reluf(apB - IC + MARGIN) + reluf(apB - VB + MARGIN)) * 0.5f;
    float cC =  reluf(apC - IC + MARGIN);
    float c_unconf = (i < 32) ? cA : ((i < 64) ? cB : cC);

    bool confident = prob[i] >= thr;
    cArr[i]    = confident ? c_conf : c_unconf;
    confArr[i] = confident ? 1 : 0;
    corrArr[i] = (confident && (an >= ap)) ? 1 : 0;
  }
}

// ---------------- kernel 5: final reduction ----------------

__global__ void __launch_bounds__(1024) k_final(const float* __restrict__ c,
                                                const int* __restrict__ conf,
                                                const int* __restrict__ corr,
                                                float* __restrict__ out) {
  __shared__ float sl[1024];
  __shared__ int   sc[1024], sn[1024];
  __shared__ int   jmax;
  int t = threadIdx.x;
  if (t == 0) {
    int j = -1;
    for (int r = 0; r < 64; ++r) if (!conf[r]) j = r;
    jmax = j;
  }
  __syncthreads();
  int j = jmax;
  float ls = 0.f; int cs = 0, ns = 0;
  for (int r = t; r < NROWS; r += 1024) {
    bool ow   = (!conf[r]) && (r < 64);
    bool keep = (r > j) && !ow;
    if (keep) ls += c[r];
    cs += corr[r]; ns += conf[r];
  }
  sl[t] = ls; sc[t] = cs; sn[t] = ns;
  __syncthreads();
  for (int s = 512; s > 0; s >>= 1) {
    if (t < s) { sl[t] += sl[t + s]; sc[t] += sc[t + s]; sn[t] += sn[t + s]; }
    __syncthreads();
  }
  if (t == 0) {
    float loss = sl[0] + ((j >= 0) ? c[j] : 0.0f);
    out[0] = loss / (float)NROWS;
    out[1] = (float)sc[0];
    out[2] = (float)sn[0];
  }
}

// ---------------- launch ----------------

extern "C" void kernel_launch(void* const* d_in, const int* in_sizes, int n_in,
                              void* d_out, int out_size, void* d_ws, size_t ws_size,
                              hipStream_t stream) {
  const float* X      = (const float*)d_in[0];   // (4096,1024) f32
  const float* P      = (const float*)d_in[1];   // (4096,64)   f32
  const int*   tgt    = (const int*)  d_in[2];   // (4096)      i32
  // d_in[3] = true_targets (unused by reference output path)
  const float* prob   = (const float*)d_in[4];   // (4096)      f32
  const float* thr    = (const float*)d_in[5];   // scalar      f32

  float* dist   = (float*)d_ws;                          // 4096*4096 f32 (64 MB)
  float* sq     = dist + (size_t)NROWS * NROWS;          // 4096
  float* cArr   = sq + NROWS;                            // 4096
  int*   pcls   = (int*)(cArr + NROWS);                  // 4096
  int*   confA  = pcls + NROWS;                          // 4096
  int*   corrA  = confA + NROWS;                         // 4096
  float* outf   = (float*)d_out;

  k_rowsq  <<<NROWS / 8, 256, 0, stream>>>(X, sq);
  k_predcls<<<NROWS / 256, 256, 0, stream>>>(P, pcls);

  const int ntile = MT * (MT + 1) / 2;   // triangular 32x32 macro-tile grid
  k_dist<<<ntile, 32, 0, stream>>>(X, sq, dist);

  k_mine<<<NROWS / 8, 256, 0, stream>>>(dist, tgt, prob, pcls, thr,
                                        cArr, confA, corrA);

  k_final<<<1, 1024, 0, stream>>>(cArr, confA, corrA, outf);
}